// EmbeddingLayerWithDropout_60009283060151
// MI455X (gfx1250) — compile-verified
//
#include <hip/hip_runtime.h>
#include <hip/hip_bf16.h>
#include <stdint.h>

// Embedding gather with word dropout: out[t,:] = W[id[t],:] * ((U[id[t],:] < 0.9)/0.9)
// B*S = 16384 tokens, D = 1024 f32.  Pure bandwidth kernel (~200 MB => ~8.6us @ 23.3 TB/s).
// CDNA5 path: TDM (tensor_load_to_lds) DMAs the 4KB weight row + 4KB mask row into LDS,
// double-buffered across tokens, synced with s_wait_tensorcnt; waves compute from LDS
// with 128-bit ds_loads and write the output with non-temporal 128-bit global stores
// (output is write-once; NT keeps L2 free for the reused weight/mask rows).

#define DIM      1024
#define KEEP     0.9f
#define NBLOCKS  2048
#define NTHREADS 256

typedef unsigned int v4u32 __attribute__((ext_vector_type(4)));
typedef int          v8i32 __attribute__((ext_vector_type(8)));
typedef int          v4i32 __attribute__((ext_vector_type(4)));
typedef float        v4f   __attribute__((ext_vector_type(4)));  // native vector: OK for NT builtins

// Issue one TDM load of a contiguous row of DIM f32 elements from global -> LDS.
// D# layout per CDNA5 ISA ch.8 (group0: count/lds_addr/global_addr/type,
// group1: data_size/tensor dims/tile dims/strides; groups 2-3 zero => <=2D tensor).
__device__ __forceinline__ void tdm_load_row_f32(uint32_t lds_byte_off, uint64_t gaddr) {
  v4u32 g0;
  g0[0] = 1u;                                                    // count=1, user descriptor
  g0[1] = lds_byte_off;                                          // D#.lds_addr (bytes)
  g0[2] = (uint32_t)(gaddr & 0xFFFFFFFFull);                     // global_addr[31:0]
  g0[3] = (uint32_t)((gaddr >> 32) & 0x1FFFFFFull) | (2u << 30); // global_addr[56:32] | type=2

  v8i32 g1;
  g1[0] = (int)(2u << 16);                      // workgroup_mask=0, data_size=2 (4 bytes)
  g1[1] = (int)((uint32_t)DIM << 16);           // tensor_dim0[15:0] @ bits 79:48 (low half)
  g1[2] = (int)(1u << 16);                      // tensor_dim0 hi = 0 | tensor_dim1 = 1
  g1[3] = (int)((uint32_t)DIM << 16);           // tensor_dim1 hi = 0 | tile_dim0 = DIM
  g1[4] = (int)1u;                              // tile_dim1 = 1, tile_dim2 = 0
  g1[5] = (int)DIM;                             // tensor_dim0_stride[31:0]
  g1[6] = 0;                                    // stride hi, tensor_dim1_stride lo
  g1[7] = 0;

  v4i32 gz = {0, 0, 0, 0};
#if defined(__clang_major__) && (__clang_major__ >= 23)
  v8i32 gz8 = {0, 0, 0, 0, 0, 0, 0, 0};
  __builtin_amdgcn_tensor_load_to_lds(g0, g1, gz, gz, gz8, 0);
#else
  __builtin_amdgcn_tensor_load_to_lds(g0, g1, gz, gz, 0);
#endif
}

__global__ __launch_bounds__(NTHREADS)
void embed_dropout_tdm(const long long* __restrict__ ids,
                       const float* __restrict__ weight,
                       const float* __restrict__ mask_u,
                       float* __restrict__ out,
                       int ntok) {
  // Two buffers, each: [0..DIM) weight row, [DIM..2*DIM) mask row.  16 KB total.
  __shared__ __align__(16) float smem[2][2 * DIM];

  const int tid    = (int)threadIdx.x;
  const int stride = (int)gridDim.x;
  const int tk0    = (int)blockIdx.x;

  const uint64_t wbase = (uint64_t)(uintptr_t)weight;
  const uint64_t mbase = (uint64_t)(uintptr_t)mask_u;

  // Generic-pointer low 32 bits == LDS byte offset (LDS aperture keeps offset in addr[31:0]).
  const uint32_t lbase = (uint32_t)(uintptr_t)&smem[0][0];

  // Uniform scalar predicate: only wave 0 issues TDM ops (TDM ignores EXEC, so an
  // EXEC-masked "if" would not stop other waves from issuing duplicate DMAs).
  const bool issuer = (__builtin_amdgcn_readfirstlane((unsigned)tid >> 5) == 0u);

  auto issue_token = [&](int tk, int buf) {
    uint32_t row = __builtin_amdgcn_readfirstlane((uint32_t)(uint64_t)ids[tk]);
    uint64_t off = (uint64_t)row * (uint64_t)(DIM * sizeof(float));
    uint32_t l   = lbase + (uint32_t)buf * (2u * DIM * (uint32_t)sizeof(float));
    tdm_load_row_f32(l, wbase + off);                                 // weight row -> LDS
    tdm_load_row_f32(l + DIM * (uint32_t)sizeof(float), mbase + off); // mask row -> LDS
  };

  if (issuer && tk0 < ntok) {
    issue_token(tk0, 0);
  }

  const float inv = 1.0f / KEEP;
  int i = 0;
  for (int tk = tk0; tk < ntok; tk += stride, ++i) {
    const int buf = i & 1;

    if (issuer) {
      const int nxt = tk + stride;
      if (nxt < ntok) {
        __builtin_prefetch(ids + nxt, 0, 1);    // global_prefetch_b8 on next id line
        issue_token(nxt, buf ^ 1);              // prefetch next token's rows
        __builtin_amdgcn_s_wait_tensorcnt(2);   // in-order: current token's 2 DMAs done
      } else {
        __builtin_amdgcn_s_wait_tensorcnt(0);   // drain tail
      }
    }
    __syncthreads();  // publish LDS tile to all 8 waves

    // 256 threads x v4f == 1024 elements: 128-bit ds_load + NT 128-bit global_store.
    const v4f w4 = ((const v4f*)&smem[buf][0])[tid];
    const v4f m4 = ((const v4f*)&smem[buf][DIM])[tid];
    v4f o;
    o.x = (m4.x < KEEP) ? w4.x * inv : 0.0f;
    o.y = (m4.y < KEEP) ? w4.y * inv : 0.0f;
    o.z = (m4.z < KEEP) ? w4.z * inv : 0.0f;
    o.w = (m4.w < KEEP) ? w4.w * inv : 0.0f;

    // Output is streamed (write-once, never re-read): non-temporal store keeps the
    // 192MB L2 reserved for the weight/mask rows, which DO have cross-token reuse.
    v4f* orow = (v4f*)(out + (size_t)tk * DIM);
    __builtin_nontemporal_store(o, &orow[tid]);

    __syncthreads();  // all reads of this buffer done before it is re-filled
  }
}

extern "C" void kernel_launch(void* const* d_in, const int* in_sizes, int n_in,
                              void* d_out, int out_size, void* d_ws, size_t ws_size,
                              hipStream_t stream) {
  const long long* ids   = (const long long*)d_in[0];  // input_ids, int64 (B*S)
  const float* weight    = (const float*)d_in[1];      // (V, D) f32
  const float* mask_u    = (const float*)d_in[2];      // (V, D) f32
  float* out             = (float*)d_out;              // (B, S, D) f32

  const int ntok = in_sizes[0];                        // B*S = 16384

  dim3 grid(NBLOCKS), block(NTHREADS);
  hipLaunchKernelGGL(embed_dropout_tdm, grid, block, 0, stream,
                     ids, weight, mask_u, out, ntok);
}